// GraphSageLayer_52664888984237
// MI455X (gfx1250) — compile-verified
//
#include <hip/hip_runtime.h>
#include <hip/hip_bf16.h>

typedef __attribute__((ext_vector_type(2))) float v2f;
typedef __attribute__((ext_vector_type(8))) float v8f;

#define D 128          // D_IN == D_OUT == 128
#define WPAD 132       // LDS row stride in floats (128 + 4 pad -> conflict-free)
#define LDS_BYTES (2 * 128 * WPAD * 4)   // two staged weight matrices, 132 KB

// ---------------------------------------------------------------------------
// Kernel 1: zero the aggregation buffer + degree buffer (grid-stride).
// ---------------------------------------------------------------------------
__global__ void sage_zero_kernel(float* __restrict__ buf, long total) {
    long i = (long)blockIdx.x * blockDim.x + threadIdx.x;
    long stride = (long)gridDim.x * blockDim.x;
    for (; i < total; i += stride) buf[i] = 0.0f;
}

// ---------------------------------------------------------------------------
// Kernel 2: edge scatter. One wave32 per edge per iteration.
// Lane l loads h[src][4l..4l+3] as float4 (512B coalesced per wave) and
// atomically adds into agg[dst]. Lane 0 increments deg[dst].
// Prefetch next edge's source row to hide gather latency behind atomics.
// ---------------------------------------------------------------------------
__global__ void sage_scatter_kernel(const float* __restrict__ h,
                                    const long long* __restrict__ edges,
                                    float* __restrict__ agg,
                                    float* __restrict__ deg,
                                    int E, int N) {
    const int lane = threadIdx.x & 31;
    const int wavesPerBlock = blockDim.x >> 5;
    int wave = blockIdx.x * wavesPerBlock + (threadIdx.x >> 5);
    const int nwaves = gridDim.x * wavesPerBlock;

    for (int e = wave; e < E; e += nwaves) {
        long long s = edges[e];        // src row
        long long d = edges[E + e];    // dst row

        // prefetch the NEXT iteration's source row (global_prefetch_b8)
        int en = e + nwaves;
        if (en < E) {
            long long sn = edges[en];
            __builtin_prefetch(h + sn * D + lane * 4, 0, 0);
        }

        const float4 v = ((const float4*)(h + s * D))[lane];
        float* ad = agg + d * D + lane * 4;
        atomicAdd(ad + 0, v.x);
        atomicAdd(ad + 1, v.y);
        atomicAdd(ad + 2, v.z);
        atomicAdd(ad + 3, v.w);
        if (lane == 0) atomicAdd(deg + d, 1.0f);
    }
}

// ---------------------------------------------------------------------------
// Kernel 3: fused dual-GEMM + bias + ReLU using V_WMMA_F32_16X16X4_F32.
//   out = relu(h @ Ws^T + b + (agg/deg) @ Wn^T)
// Block = 256 threads = 8 waves; each wave computes a 16-row x 128-col stripe
// (8 accumulator tiles). Both weight matrices are staged into LDS once per
// block (row-padded to 132 floats for bank-conflict-free ds_load_b64), then
// the k-loop reads B fragments from LDS in batched register arrays so the
// 16 WMMAs per k-step issue back-to-back instead of load/wait/wmma lockstep.
//
// f32 A (16x4) layout: lanes 0-15 hold M=lane, VGPR0=K0,VGPR1=K1;
//                      lanes 16-31 hold M=lane-16, VGPR0=K2,VGPR1=K3.
// B (4x16) mirrors this with N on lanes. C/D: VGPR r -> M = r + (lane>=16?8:0).
// ---------------------------------------------------------------------------
__global__ __launch_bounds__(256) void sage_gemm_kernel(
        const float* __restrict__ h,
        const float* __restrict__ agg,
        const float* __restrict__ deg,
        const float* __restrict__ Wself,
        const float* __restrict__ bself,
        const float* __restrict__ Wneigh,
        float* __restrict__ out,
        int N) {
    extern __shared__ float lds[];                 // [2][128][WPAD]
    float* ldsS = lds;                             // staged W_self
    float* ldsN = lds + 128 * WPAD;                // staged W_neigh

    // --- cooperative staging of both weight matrices (all waves) ---
    // 128 rows x 32 float4 per matrix = 4096 float4 chunks.
    for (int i = threadIdx.x; i < 128 * 32; i += 256) {
        const int r = i >> 5;            // row (output channel j)
        const int c = (i & 31) << 2;     // column (k), float4 granularity
        const float4 vs = ((const float4*)Wself)[i];
        const float4 vn = ((const float4*)Wneigh)[i];
        *(float4*)&ldsS[r * WPAD + c] = vs;
        *(float4*)&ldsN[r * WPAD + c] = vn;
    }
    __syncthreads();

    const int wave = threadIdx.x >> 5;
    const int lane = threadIdx.x & 31;
    const int rowTile = blockIdx.x * 8 + wave;
    const int row0 = rowTile * 16;
    if (row0 >= N) return;                 // wave-uniform: EXEC stays full

    const int idx  = lane & 15;            // M index for A, N index for B
    const int koff = (lane >> 4) << 1;     // 0 for lanes 0-15, 2 for 16-31

    const int rA = min(row0 + idx, N - 1);
    const float* hrow = h   + (long)rA * D;
    const float* arow = agg + (long)rA * D;
    const float invd = 1.0f / fmaxf(deg[rA], 1.0f);

    v8f acc[8] = {};                       // 8 tiles x 8 VGPRs = 64 VGPRs

    for (int k0 = 0; k0 < D; k0 += 4) {
        // A fragments (global; rows stream sequentially, L0-resident)
        v2f a_h, a_n;
        a_h.x = hrow[k0 + koff];
        a_h.y = hrow[k0 + koff + 1];
        a_n.x = arow[k0 + koff] * invd;
        a_n.y = arow[k0 + koff + 1] * invd;

        // B fragments for all 8 column tiles, batched from LDS
        v2f bs[8], bn[8];
#pragma unroll
        for (int ct = 0; ct < 8; ct++) {
            const int brow = ct * 16 + idx;
            bs[ct] = *(const v2f*)&ldsS[brow * WPAD + k0 + koff];
            bn[ct] = *(const v2f*)&ldsN[brow * WPAD + k0 + koff];
        }
#pragma unroll
        for (int ct = 0; ct < 8; ct++) {
            acc[ct] = __builtin_amdgcn_wmma_f32_16x16x4_f32(
                false, a_h, false, bs[ct], (short)0, acc[ct], false, false);
            acc[ct] = __builtin_amdgcn_wmma_f32_16x16x4_f32(
                false, a_n, false, bn[ct], (short)0, acc[ct], false, false);
        }
    }

    // Epilogue: bias + ReLU, store per documented C/D VGPR layout.
    const int mhi = (lane >> 4) * 8;
#pragma unroll
    for (int ct = 0; ct < 8; ct++) {
        const float bn_ = bself[ct * 16 + idx];
#pragma unroll
        for (int r = 0; r < 8; r++) {
            const int m = row0 + r + mhi;
            if (m < N) {
                float v = acc[ct][r] + bn_;
                out[(long)m * D + ct * 16 + idx] = fmaxf(v, 0.0f);
            }
        }
    }
}

// ---------------------------------------------------------------------------
// Host launcher
// ---------------------------------------------------------------------------
extern "C" void kernel_launch(void* const* d_in, const int* in_sizes, int n_in,
                              void* d_out, int out_size, void* d_ws, size_t ws_size,
                              hipStream_t stream) {
    const float*     h      = (const float*)d_in[0];
    const long long* edges  = (const long long*)d_in[1];   // int64 [2, E]
    const float*     Wself  = (const float*)d_in[2];
    const float*     bself  = (const float*)d_in[3];
    const float*     Wneigh = (const float*)d_in[4];
    float*           out    = (float*)d_out;

    const int N = in_sizes[0] / D;
    const int E = in_sizes[1] / 2;

    float* agg = (float*)d_ws;               // N*D floats
    float* deg = agg + (size_t)N * D;        // N floats

    // 1) zero agg + deg (contiguous in workspace)
    const long total = (long)N * D + N;
    sage_zero_kernel<<<4096, 256, 0, stream>>>(agg, total);

    // 2) scatter-add over edges (8 waves/block, grid-stride over edges)
    sage_scatter_kernel<<<4096, 256, 0, stream>>>(h, edges, agg, deg, E, N);

    // 3) fused dual WMMA-GEMM + bias + ReLU (8 waves/block = 128 rows/block)
    const int blocks = (N + 127) / 128;
    sage_gemm_kernel<<<blocks, 256, LDS_BYTES, stream>>>(h, agg, deg, Wself,
                                                         bself, Wneigh, out, N);
}